// FracAdaptFilter_24885040513287
// MI455X (gfx1250) — compile-verified
//
#include <hip/hip_runtime.h>
#include <hip/hip_bf16.h>

#define NN 50000
#define EE 600000
#define NFEAT 128
#define KK 15
#define DRUG_NUM 10000
#define FEAT_PROJ 16
#define HYPER_H 32

typedef __attribute__((ext_vector_type(2))) float v2f;
typedef __attribute__((ext_vector_type(8))) float v8f;

// ---------------------------------------------------------------------------
// deg[i] = sum of edge_vals over edges with row == i
// ---------------------------------------------------------------------------
__global__ __launch_bounds__(256) void deg_kernel(const int* __restrict__ rows,
                                                  const float* __restrict__ vals,
                                                  float* __restrict__ deg) {
    int e = blockIdx.x * blockDim.x + threadIdx.x;
    if (e >= EE) return;
    unsafeAtomicAdd(&deg[rows[e]], vals[e]);
}

// ---------------------------------------------------------------------------
// feat_low = x @ W_fp + b_fp  via V_WMMA_F32_16X16X4_F32
// One wave per 16-row tile; 32 WMMA steps over K=128.
// ---------------------------------------------------------------------------
__global__ __launch_bounds__(256) void feat_proj_wmma(const float* __restrict__ x,
                                                      const float* __restrict__ Wfp,
                                                      const float* __restrict__ bfp,
                                                      float* __restrict__ feat_low) {
    const int lane = threadIdx.x & 31;
    const int wave = blockIdx.x * (blockDim.x >> 5) + (threadIdx.x >> 5);
    const int tile = wave;                       // 16 rows per tile, 3125 tiles
    if (tile >= NN / 16) return;                 // wave-uniform guard (EXEC all-1 inside)

    const int half = lane >> 4;                  // 0: K pair {0,1}; 1: K pair {2,3}
    const int l15  = lane & 15;
    const size_t arow = (size_t)(tile * 16 + l15) * NFEAT;

    v8f acc = {};
#pragma unroll
    for (int kk = 0; kk < NFEAT / 4; ++kk) {
        const int k0 = kk * 4 + half * 2;
        v2f a, b;
        a.x = x[arow + k0];
        a.y = x[arow + k0 + 1];
        b.x = Wfp[(size_t)(k0    ) * FEAT_PROJ + l15];
        b.y = Wfp[(size_t)(k0 + 1) * FEAT_PROJ + l15];
        // 8 args: (neg_a, A, neg_b, B, c_mod, C, reuse_a, reuse_b)
        acc = __builtin_amdgcn_wmma_f32_16x16x4_f32(false, a, false, b,
                                                    (short)0, acc, false, false);
    }

    const float bias = bfp[l15];
#pragma unroll
    for (int r = 0; r < 8; ++r) {
        int m = tile * 16 + r + half * 8;        // C/D layout: VGPR r, halves
        feat_low[(size_t)m * FEAT_PROJ + l15] = acc[r] + bias;
    }
}

// ---------------------------------------------------------------------------
// Per-node: ctx -> MLP -> alpha,t -> fractional coefficients
// ---------------------------------------------------------------------------
__device__ __forceinline__ float softplus_f(float v) {
    return fmaxf(v, 0.0f) + log1pf(expf(-fabsf(v)));
}

__global__ __launch_bounds__(256) void mlp_coeffs_kernel(
        const float* __restrict__ feat_low, const float* __restrict__ deg,
        const float* __restrict__ W1, const float* __restrict__ b1,
        const float* __restrict__ W2, const float* __restrict__ b2,
        const float* __restrict__ ab_drug, const float* __restrict__ ab_prot,
        const float* __restrict__ t_bias,
        float* __restrict__ coeffs, float* __restrict__ alpha_out,
        float* __restrict__ t_out) {
    __shared__ float sW1[18 * HYPER_H];
    __shared__ float sb1[HYPER_H];
    __shared__ float sW2[HYPER_H * 2];
    __shared__ float sb2[2];
    for (int j = threadIdx.x; j < 18 * HYPER_H; j += blockDim.x) sW1[j] = W1[j];
    for (int j = threadIdx.x; j < HYPER_H; j += blockDim.x)      sb1[j] = b1[j];
    for (int j = threadIdx.x; j < HYPER_H * 2; j += blockDim.x)  sW2[j] = W2[j];
    if (threadIdx.x < 2) sb2[threadIdx.x] = b2[threadIdx.x];
    __syncthreads();

    int i = blockIdx.x * blockDim.x + threadIdx.x;
    if (i >= NN) return;

    float ctx[18];
#pragma unroll
    for (int j = 0; j < FEAT_PROJ; ++j) ctx[j] = feat_low[(size_t)i * FEAT_PROJ + j];
    ctx[16] = log1pf(deg[i]);
    ctx[17] = (i >= DRUG_NUM) ? 1.0f : 0.0f;

    float raw0 = sb2[0], raw1 = sb2[1];
#pragma unroll 4
    for (int h = 0; h < HYPER_H; ++h) {
        float acc = sb1[h];
#pragma unroll
        for (int j = 0; j < 18; ++j) acc = fmaf(ctx[j], sW1[j * HYPER_H + h], acc);
        acc = fmaxf(acc, 0.0f);
        raw0 = fmaf(acc, sW2[h * 2 + 0], raw0);
        raw1 = fmaf(acc, sW2[h * 2 + 1], raw1);
    }

    float ab = (i < DRUG_NUM) ? ab_drug[0] : ab_prot[0];
    float alpha = softplus_f(raw0 + ab) + 0.05f;
    float t     = softplus_f(raw1 + t_bias[0]) + 0.01f;
    alpha_out[i] = alpha;
    t_out[i]     = t;

    // clipped copies used only for the coefficients (matches reference)
    float ac = fminf(fmaxf(alpha, 0.05f), 3.0f);
    float tc = fminf(fmaxf(t, 0.01f), 10.0f);
    float s    = tc / (1.0f + tc);
    float base = expf(-ac * log1pf(tc));          // (1+t)^(-alpha)
    coeffs[i] = base;
    float rising = 1.0f, s_pow = 1.0f;
#pragma unroll
    for (int k = 1; k <= KK; ++k) {
        rising *= (ac + (float)k - 1.0f) / (float)k;
        s_pow  *= s;
        coeffs[(size_t)k * NN + i] = base * rising * s_pow;
    }
}

// ---------------------------------------------------------------------------
// out[i,:] += c[i] * power[i,:]  AND zero the next power buffer in the same
// pass (saves a dedicated 25.6MB memset sweep per filter iteration).
// ---------------------------------------------------------------------------
__global__ __launch_bounds__(256) void accum_zero_kernel(
        const float4* __restrict__ power, const float* __restrict__ ck,
        float4* __restrict__ out, float4* __restrict__ zero_buf) {
    int idx = blockIdx.x * blockDim.x + threadIdx.x;   // over N*32 float4s
    if (idx >= NN * (NFEAT / 4)) return;
    int node = idx >> 5;                               // 32 float4 per row
    float c = ck[node];
    float4 p = power[idx];
    float4 o = out[idx];
    o.x = fmaf(c, p.x, o.x);
    o.y = fmaf(c, p.y, o.y);
    o.z = fmaf(c, p.z, o.z);
    o.w = fmaf(c, p.w, o.w);
    out[idx] = o;
    zero_buf[idx] = make_float4(0.f, 0.f, 0.f, 0.f);
}

// ---------------------------------------------------------------------------
// newPower[row,:] += val * power[col,:]  — one wave per edge, float4 per lane.
// Edge index forced wave-uniform -> row/col/val become SALU (s_load) reads;
// gather is a fully coalesced 512B row; scatter uses hardware f32 atomics.
// ---------------------------------------------------------------------------
__global__ __launch_bounds__(256) void spmm_kernel(const int* __restrict__ rows,
                                                   const int* __restrict__ cols,
                                                   const float* __restrict__ vals,
                                                   const float* __restrict__ power,
                                                   float* __restrict__ newPower) {
    const int lane = threadIdx.x & 31;
    int e = blockIdx.x * (blockDim.x >> 5) + (threadIdx.x >> 5);
    e = __builtin_amdgcn_readfirstlane(e);             // provably wave-uniform
    if (e >= EE) return;
    const int   r = rows[e];
    const int   c = cols[e];
    const float v = vals[e];

    const float4 p = ((const float4*)(power + (size_t)c * NFEAT))[lane];
    float* dst = newPower + (size_t)r * NFEAT + lane * 4;
    unsafeAtomicAdd(dst + 0, v * p.x);
    unsafeAtomicAdd(dst + 1, v * p.y);
    unsafeAtomicAdd(dst + 2, v * p.z);
    unsafeAtomicAdd(dst + 3, v * p.w);
}

// ---------------------------------------------------------------------------
extern "C" void kernel_launch(void* const* d_in, const int* in_sizes, int n_in,
                              void* d_out, int out_size, void* d_ws, size_t ws_size,
                              hipStream_t stream) {
    const float* x       = (const float*)d_in[0];
    const int*   e_rows  = (const int*)  d_in[1];
    const int*   e_cols  = (const int*)  d_in[2];
    const float* e_vals  = (const float*)d_in[3];
    const float* W_fp    = (const float*)d_in[4];
    const float* b_fp    = (const float*)d_in[5];
    const float* W1      = (const float*)d_in[6];
    const float* b1      = (const float*)d_in[7];
    const float* W2      = (const float*)d_in[8];
    const float* b2      = (const float*)d_in[9];
    const float* ab_drug = (const float*)d_in[10];
    const float* ab_prot = (const float*)d_in[11];
    const float* t_bias  = (const float*)d_in[12];

    float* out       = (float*)d_out;            // [N,128]
    float* alpha_out = out + (size_t)NN * NFEAT; // [N]
    float* t_out     = alpha_out + NN;           // [N]

    // workspace layout (floats): deg | feat_low(16N) | coeffs(16N) | pA(128N) | pB(128N)
    float* ws       = (float*)d_ws;
    float* deg      = ws;
    float* feat_low = ws + (size_t)NN;
    float* coeffs   = ws + (size_t)17 * NN;
    float* pA       = ws + (size_t)33 * NN;
    float* pB       = ws + (size_t)161 * NN;
    const size_t mat_bytes = (size_t)NN * NFEAT * sizeof(float);

    // 1) degree (row sums)
    hipMemsetAsync(deg, 0, (size_t)NN * sizeof(float), stream);
    deg_kernel<<<(EE + 255) / 256, 256, 0, stream>>>(e_rows, e_vals, deg);

    // 2) feat_proj via WMMA f32 16x16x4
    {
        int tiles = NN / 16;                     // 3125
        int waves_per_block = 256 / 32;          // 8
        int blocks = (tiles + waves_per_block - 1) / waves_per_block;
        feat_proj_wmma<<<blocks, 256, 0, stream>>>(x, W_fp, b_fp, feat_low);
    }

    // 3) hyper MLP + alpha/t + fractional coefficients
    mlp_coeffs_kernel<<<(NN + 255) / 256, 256, 0, stream>>>(
        feat_low, deg, W1, b1, W2, b2, ab_drug, ab_prot, t_bias,
        coeffs, alpha_out, t_out);

    // 4) power = x ; out = 0
    hipMemcpyAsync(pA, x, mat_bytes, hipMemcpyDeviceToDevice, stream);
    hipMemsetAsync(out, 0, mat_bytes, stream);

    const int accum_blocks = (NN * (NFEAT / 4) + 255) / 256;      // 6250
    const int spmm_blocks  = (EE + 7) / 8;                         // 8 edges/block

    // 5) polynomial filter: out += c_k * power; power = A @ power
    for (int k = 0; k < KK; ++k) {
        // accum also zeroes pB so the following scatter-add starts from 0
        accum_zero_kernel<<<accum_blocks, 256, 0, stream>>>(
            (const float4*)pA, coeffs + (size_t)k * NN, (float4*)out, (float4*)pB);
        spmm_kernel<<<spmm_blocks, 256, 0, stream>>>(e_rows, e_cols, e_vals, pA, pB);
        float* tmp = pA; pA = pB; pB = tmp;
    }
    // final k == K term (zero target unused afterwards; harmless)
    accum_zero_kernel<<<accum_blocks, 256, 0, stream>>>(
        (const float4*)pA, coeffs + (size_t)KK * NN, (float4*)out, (float4*)pB);
}